// FeatSimLoss_41472204210804
// MI455X (gfx1250) — compile-verified
//
#include <hip/hip_runtime.h>

// FeatSimLoss for MI455X (gfx1250, wave32).
//
// Pipeline (3 kernels, all deterministic, no atomics):
//  A) feat_gram_kernel: 64x64-grid norm + 4 neighbor-dot tables via
//     V_WMMA_F32_16X16X4_F32 Gram chains (fp32-exact, 5 chains x 64 steps).
//  B) main_kernel: per 128-res pixel — in-tile softmax into LDS, pos/neg via
//     19-dim prob dots, 9 sims reconstructed from the WMMA tables, top-k with
//     jax.lax.top_k tie semantics, block-reduced partial sums + state_sim.
//  C) finalize_kernel: fixed-order reduction of 128 partial triples -> the two
//     scalar losses.

#define H128 128
#define W128 128
#define H64  64
#define W64  64
#define CF   256
#define CS   19
#define NB   2

#define TAB_PLANE (H64 * W64)          // 4096 floats per (table, batch) plane
#define TAB_SZ    (5 * NB * TAB_PLANE) // 40960 floats of tables in ws
#define NPART     128                  // main_kernel block count

typedef float v2f __attribute__((ext_vector_type(2)));
typedef float v8f __attribute__((ext_vector_type(8)));

// ---------------------------------------------------------------------------
// Kernel A: per (b, row y, 16-cell strip) build N / dotE / dotS / dotSE / dotSW
// tables with fp32 WMMA Gram chains. One wave per block; EXEC is all-ones
// (required for WMMA). Diagonals extracted through LDS.
// ---------------------------------------------------------------------------
__global__ __launch_bounds__(32) void feat_gram_kernel(
    const float* __restrict__ F,   // [2,256,64,64]
    float* __restrict__ tab) {     // ws: 5 tables [2][64][64]
  const int x0   = blockIdx.x * 16;
  const int y    = blockIdx.y;
  const int b    = blockIdx.z;
  const int lane = threadIdx.x;
  const int lm   = lane & 15;
  const int half = lane >> 4;
  const int y1   = min(y + 1, H64 - 1);

  // Lane-private cell coordinates for the 5 fragment sources. Clamped edges
  // produce table entries that are never read by kernel B (parity analysis).
  const int xA  = x0 + lm;
  const int xE  = min(xA + 1, W64 - 1);
  const int xSW = max(xA - 1, 0);

  const size_t plane = (size_t)H64 * W64;
  const size_t bBase = (size_t)b * CF * plane;
  const size_t offA  = bBase + (size_t)y  * W64 + xA;
  const size_t offE  = bBase + (size_t)y  * W64 + xE;
  const size_t offS  = bBase + (size_t)y1 * W64 + xA;
  const size_t offSE = bBase + (size_t)y1 * W64 + xE;
  const size_t offSW = bBase + (size_t)y1 * W64 + xSW;

  v8f accN = {}, accE = {}, accS = {}, accSE = {}, accSW = {};

  for (int k = 0; k < CF / 4; ++k) {
    // 16x4 fp32 A-frag layout: lanes 0-15 hold (M=lane, K=0/1) in v0/v1,
    // lanes 16-31 hold (M=lane-16, K=2/3). B mirrors with N=lane, so the
    // per-lane data is identical -> one 2-dword load pattern per fragment.
    const int    c0  = 4 * k + 2 * half;
    const size_t co0 = (size_t)c0 * plane;
    const size_t co1 = co0 + plane;
    v2f a, bE, bS, bSE, bSW;
    a[0]   = F[offA  + co0]; a[1]   = F[offA  + co1];
    bE[0]  = F[offE  + co0]; bE[1]  = F[offE  + co1];
    bS[0]  = F[offS  + co0]; bS[1]  = F[offS  + co1];
    bSE[0] = F[offSE + co0]; bSE[1] = F[offSE + co1];
    bSW[0] = F[offSW + co0]; bSW[1] = F[offSW + co1];

    accN  = __builtin_amdgcn_wmma_f32_16x16x4_f32(false, a, false, a,   (short)0, accN,  false, false);
    accE  = __builtin_amdgcn_wmma_f32_16x16x4_f32(false, a, false, bE,  (short)0, accE,  false, false);
    accS  = __builtin_amdgcn_wmma_f32_16x16x4_f32(false, a, false, bS,  (short)0, accS,  false, false);
    accSE = __builtin_amdgcn_wmma_f32_16x16x4_f32(false, a, false, bSE, (short)0, accSE, false, false);
    accSW = __builtin_amdgcn_wmma_f32_16x16x4_f32(false, a, false, bSW, (short)0, accSW, false, false);
  }

  // Extract diagonals: C/D element (M,N) lives in VGPR M%8, lane (M>=8)*16 + N.
  __shared__ float tile[5][16 * 16];
#pragma unroll
  for (int v = 0; v < 8; ++v) {
    const int row = v + 8 * half;
    tile[0][row * 16 + lm] = accN[v];
    tile[1][row * 16 + lm] = accE[v];
    tile[2][row * 16 + lm] = accS[v];
    tile[3][row * 16 + lm] = accSE[v];
    tile[4][row * 16 + lm] = accSW[v];
  }
  __syncthreads();
  if (lane < 16) {
    const size_t o = (size_t)b * TAB_PLANE + (size_t)y * W64 + (x0 + lane);
#pragma unroll
    for (int t = 0; t < 5; ++t)
      tab[(size_t)t * NB * TAB_PLANE + o] = tile[t][lane * 17];  // [i][i]
  }
}

// ---------------------------------------------------------------------------
// Kernel B: one thread per 128-res pixel; 16x16 pixel tile per block.
// ---------------------------------------------------------------------------
__global__ __launch_bounds__(256) void main_kernel(
    const float* __restrict__ F,        // [2,256,64,64]
    const float* __restrict__ logits,   // [2,19,128,128]
    const float* __restrict__ tab,      // ws tables
    float* __restrict__ out_state,      // d_out + 2, [2,128,128]
    float* __restrict__ partials) {     // ws: 128 * 3 floats
  __shared__ float probT[18 * 18][CS + 1];  // [.. ][19] holds psum
  __shared__ float red[3][256];

  const int tid = threadIdx.y * 16 + threadIdx.x;
  const int b   = blockIdx.z;
  const int h0  = blockIdx.y * 16, w0 = blockIdx.x * 16;

  // Stage softmax(prob) + psum for the 18x18 halo tile; OOB halo -> zeros,
  // which makes pos/neg vanish for padded neighbors exactly like the ref.
  for (int idx = tid; idx < 18 * 18; idx += 256) {
    const int py = idx / 18, px = idx % 18;
    const int gh = h0 + py - 1, gw = w0 + px - 1;
    float p[CS];
    float s = 0.f;
    if (gh >= 0 && gh < H128 && gw >= 0 && gw < W128) {
      const size_t base =
          (size_t)b * CS * H128 * W128 + (size_t)gh * W128 + gw;
      float mx = -3.0e38f;
#pragma unroll
      for (int c = 0; c < CS; ++c) {
        p[c] = logits[base + (size_t)c * H128 * W128];
        mx   = fmaxf(mx, p[c]);
      }
      float se = 0.f;
#pragma unroll
      for (int c = 0; c < CS; ++c) { p[c] = __expf(p[c] - mx); se += p[c]; }
      const float inv = 1.f / se;
#pragma unroll
      for (int c = 0; c < CS; ++c) { p[c] *= inv; s += p[c]; }
    } else {
#pragma unroll
      for (int c = 0; c < CS; ++c) p[c] = 0.f;
    }
#pragma unroll
    for (int c = 0; c < CS; ++c) probT[idx][c] = p[c];
    probT[idx][CS] = s;
  }
  __syncthreads();

  const int tx = threadIdx.x, ty = threadIdx.y;
  const int h = h0 + ty, w = w0 + tx;
  const int yc = h >> 1, xc = w >> 1;

  const float* pc  = probT[(ty + 1) * 18 + (tx + 1)];
  const float  psc = pc[CS];
  const size_t tb  = (size_t)b * TAB_PLANE;
  const float  Nc  = tab[tb + yc * W64 + xc];  // table 0 = norms

  float pos[9], neg[9], sim[9];
  float simsum = 0.f;
#pragma unroll
  for (int k = 0; k < 9; ++k) {
    const int dh = k / 3 - 1, dw = k % 3 - 1;
    const int hh = h + dh, ww = w + dw;
    const float* pn = probT[(ty + 1 + dh) * 18 + (tx + 1 + dw)];
    float dp = 0.f;
#pragma unroll
    for (int c = 0; c < CS; ++c) dp += pc[c] * pn[c];
    pos[k] = dp;
    neg[k] = psc * pn[CS] - dp;

    float dist;
    if (hh < 0 || hh >= H128 || ww < 0 || ww >= W128) {
      dist = Nc;  // zero-padded neighbor: ||f||^2
    } else {
      const int gy = (hh >> 1) - yc;
      const int gx = (ww >> 1) - xc;
      if (gy == 0 && gx == 0) {
        dist = 0.f;  // same 64-grid cell after nearest upsample
      } else {
        float dot;
        if (gy == 0)
          dot = (gx > 0) ? tab[1 * NB * TAB_PLANE + tb + yc * W64 + xc]
                         : tab[1 * NB * TAB_PLANE + tb + yc * W64 + (xc - 1)];
        else if (gx == 0)
          dot = (gy > 0) ? tab[2 * NB * TAB_PLANE + tb + yc * W64 + xc]
                         : tab[2 * NB * TAB_PLANE + tb + (yc - 1) * W64 + xc];
        else if (gy > 0)
          dot = (gx > 0) ? tab[3 * NB * TAB_PLANE + tb + yc * W64 + xc]
                         : tab[4 * NB * TAB_PLANE + tb + yc * W64 + xc];
        else
          dot = (gx < 0)
                    ? tab[3 * NB * TAB_PLANE + tb + (yc - 1) * W64 + (xc - 1)]
                    : tab[4 * NB * TAB_PLANE + tb + (yc - 1) * W64 + (xc + 1)];
        const float Nn = tab[tb + (yc + gy) * W64 + (xc + gx)];
        dist = Nc + Nn - 2.f * dot;
      }
    }
    sim[k] = __expf(-dist);
    simsum += sim[k];
  }
  out_state[((size_t)b * H128 + h) * W128 + w] = simsum * (1.f / 9.f);

  // top-5 largest sims (jax.lax.top_k tie rule: lowest index wins)
  float s_pos = 0.f;
  {
    float v[9];
#pragma unroll
    for (int k = 0; k < 9; ++k) v[k] = sim[k];
#pragma unroll
    for (int t = 0; t < 5; ++t) {
      int bi = 0;
      float bv = v[0];
#pragma unroll
      for (int k = 1; k < 9; ++k)
        if (v[k] > bv) { bv = v[k]; bi = k; }
      s_pos += bv * (-pos[bi]);
      v[bi] = -3.0e38f;
    }
  }
  // top-4 smallest sims (top_k of -sim, same tie rule)
  float s_neg = 0.f;
  {
    float v[9];
#pragma unroll
    for (int k = 0; k < 9; ++k) v[k] = sim[k];
#pragma unroll
    for (int t = 0; t < 4; ++t) {
      int bi = 0;
      float bv = v[0];
#pragma unroll
      for (int k = 1; k < 9; ++k)
        if (v[k] < bv) { bv = v[k]; bi = k; }
      s_neg += (1.f - bv) * (-neg[bi]);
      v[bi] = 3.0e38f;
    }
  }

  const float f0 = F[(size_t)b * CF * H64 * W64 + (size_t)yc * W64 + xc];
  const float m  = (f0 > 0.f) ? 1.f : 0.f;

  red[0][tid] = m * s_pos;
  red[1][tid] = m * s_neg;
  red[2][tid] = m;
  __syncthreads();
  for (int off = 128; off > 0; off >>= 1) {
    if (tid < off) {
      red[0][tid] += red[0][tid + off];
      red[1][tid] += red[1][tid + off];
      red[2][tid] += red[2][tid + off];
    }
    __syncthreads();
  }
  if (tid == 0) {
    const int blockId =
        (blockIdx.z * gridDim.y + blockIdx.y) * gridDim.x + blockIdx.x;
    partials[3 * blockId + 0] = red[0][0];
    partials[3 * blockId + 1] = red[1][0];
    partials[3 * blockId + 2] = red[2][0];
  }
}

// ---------------------------------------------------------------------------
// Kernel C: fixed-order final reduction -> the two scalar losses.
// ---------------------------------------------------------------------------
__global__ __launch_bounds__(128) void finalize_kernel(
    const float* __restrict__ partials, float* __restrict__ out) {
  __shared__ float red[3][NPART];
  const int t = threadIdx.x;
  red[0][t] = partials[3 * t + 0];
  red[1][t] = partials[3 * t + 1];
  red[2][t] = partials[3 * t + 2];
  __syncthreads();
  for (int off = NPART / 2; off > 0; off >>= 1) {
    if (t < off) {
      red[0][t] += red[0][t + off];
      red[1][t] += red[1][t + off];
      red[2][t] += red[2][t + off];
    }
    __syncthreads();
  }
  if (t == 0) {
    const float cnt = red[2][0];
    out[0] = red[0][0] / (cnt * 5.0f);  // W_POS = 1
    out[1] = red[1][0] / (cnt * 4.0f);  // W_NEG = 1
  }
}

extern "C" void kernel_launch(void* const* d_in, const int* in_sizes, int n_in,
                              void* d_out, int out_size, void* d_ws,
                              size_t ws_size, hipStream_t stream) {
  (void)in_sizes; (void)n_in; (void)out_size; (void)ws_size;
  const float* feats  = (const float*)d_in[0];  // [2,256,64,64]
  const float* logits = (const float*)d_in[1];  // [2,19,128,128]
  float* out = (float*)d_out;                   // [2 scalars][2*128*128 state]
  float* ws  = (float*)d_ws;
  float* tab      = ws;            // 5 * 2 * 4096 floats
  float* partials = ws + TAB_SZ;   // 128 * 3 floats

  feat_gram_kernel<<<dim3(4, 64, 2), dim3(32), 0, stream>>>(feats, tab);
  main_kernel<<<dim3(8, 8, 2), dim3(16, 16), 0, stream>>>(
      feats, logits, tab, out + 2, partials);
  finalize_kernel<<<1, 128, 0, stream>>>(partials, out);
}